// UpperBitBoundQuantizer_attn_61718680043577
// MI455X (gfx1250) — compile-verified
//
#include <hip/hip_runtime.h>
#include <stdint.h>

// ---------------- problem constants ----------------
#define NB 8
#define NH 16
#define NT 2048
#define NC 128
#define CROW (NH*NC)            // 2048 elements per (b,t) row
#define NROWS (NB*NT)           // 16384 rows
#define NTOTAL 33554432LL       // 8*16*2048*128
#define K_IDX 16777215ULL       // floor(0.5*(N-1)) for quantile method='lower'
#define THRESH 2.0118110236220472f   // (2^9-1)/(2^8-2) = 511/254
#define CNT_FLOOR 204.8f        // C * COCO = 2048 * 0.1
#define TPB 256
#define GRID1 2048

// ---------------- workspace layout (bytes) ----------------
#define OFF_PSUM   0          // double[2048]
#define OFF_PSQ    16384      // double[2048]
#define OFF_H1     32768      // u32[2048]
#define OFF_H2     40960      // u32[2048]
#define OFF_H3     49152      // u32[1024]
#define OFF_SCAL   53248      // Scalars
#define OFF_RERR   65536      // f32[16384]
#define OFF_RASTD  131072     // f32[16384]
#define OFF_RMIN   196608     // f32[16384]
#define OFF_RMAX   262144     // f32[16384]
#define OFF_RBIT   327680     // i32[16384]

struct Scalars {
  unsigned minKey, maxKey;
  unsigned selPrefix, pad0;
  unsigned long long kRemain;
  float stdv, coeff, bandC, mean;
};

// order-preserving float<->uint key (ascending)
__device__ __forceinline__ unsigned f2key(float f){
  unsigned u = __float_as_uint(f);
  return (u & 0x80000000u) ? ~u : (u | 0x80000000u);
}
__device__ __forceinline__ float key2f(unsigned k){
  unsigned u = (k & 0x80000000u) ? (k ^ 0x80000000u) : ~k;
  return __uint_as_float(u);
}

// ---------------- CDNA5 Tensor Data Mover: 2-D fp32 tile -> LDS ----------------
typedef uint32_t u32x4 __attribute__((ext_vector_type(4)));
typedef uint32_t u32x8 __attribute__((ext_vector_type(8)));

// D# per CDNA5 ISA (groups 2/3 omitted -> <=2D tensor)
__device__ __forceinline__ void tdm_load_tile_f32(uint32_t lds_byte_addr,
                                                  const float* gptr,
                                                  uint32_t tile_x, uint32_t tile_y,
                                                  uint64_t row_stride_elems) {
  uint64_t ga = (uint64_t)(uintptr_t)gptr;
  const uint32_t td0 = 0x7FFFFFFFu;   // tensor_dim0 (huge: never OOB)
  const uint32_t td1 = 0x7FFFFFFFu;   // tensor_dim1
  u32x4 g0;
  g0[0] = 1u;                                               // count=1, user desc
  g0[1] = lds_byte_addr;                                    // lds_addr [63:32]
  g0[2] = (uint32_t)ga;                                     // global_addr[31:0]
  g0[3] = ((uint32_t)(ga >> 32) & 0x01FFFFFFu) | (2u << 30);// addr[56:32] | type=2
  u32x8 g1;
  g1[0] = (2u << 16);                                       // wg_mask=0, data_size=4B
  g1[1] = (td0 & 0xFFFFu) << 16;                            // tensor_dim0[15:0]
  g1[2] = (td0 >> 16) | ((td1 & 0xFFFFu) << 16);            // dim0 hi | dim1 lo
  g1[3] = (td1 >> 16) | ((tile_x & 0xFFFFu) << 16);         // dim1 hi | tile_dim0
  g1[4] = (tile_y & 0xFFFFu);                               // tile_dim1 | tile_dim2=0
  g1[5] = (uint32_t)row_stride_elems;                       // dim0_stride[31:0]
  g1[6] = (uint32_t)(row_stride_elems >> 32) & 0xFFFFu;     // dim0_stride[47:32]
  g1[7] = 0;
  asm volatile("tensor_load_to_lds %0, %1\n\t"
               "s_wait_tensorcnt 0x0"
               :: "s"(g0), "s"(g1) : "memory");
}

// ---------------- K0: init workspace ----------------
__global__ void k_init(char* __restrict__ ws){
  int i = blockIdx.x * blockDim.x + threadIdx.x;
  unsigned* h = (unsigned*)(ws + OFF_H1);
  if (i < 5120) h[i] = 0;                 // hist1+hist2+hist3
  if (i == 0){
    Scalars* sc = (Scalars*)(ws + OFF_SCAL);
    sc->minKey = 0xFFFFFFFFu; sc->maxKey = 0u;
    sc->selPrefix = 0u; sc->pad0 = 0u;
    sc->kRemain = K_IDX;
    sc->stdv = 0.f; sc->coeff = 0.f; sc->bandC = 0.f; sc->mean = 0.f;
  }
}

// ---------------- K1: global sum/sumsq/min/max + histogram of key[31:21] ----------------
__global__ void __launch_bounds__(TPB) k_stats_hist1(const float* __restrict__ x, char* __restrict__ ws){
  __shared__ unsigned lh[2048];
  __shared__ double sS[TPB], sQ[TPB];
  __shared__ float sMn[TPB], sMx[TPB];
  const int tid = threadIdx.x;
  for (int i = tid; i < 2048; i += TPB) lh[i] = 0u;
  __syncthreads();

  double s = 0.0, q = 0.0;
  float mn = __int_as_float(0x7f800000), mx = -mn;
  const float4* x4 = (const float4*)x;
  const long long n4 = NTOTAL >> 2;
  for (long long i = (long long)blockIdx.x * TPB + tid; i < n4; i += (long long)gridDim.x * TPB){
    float4 v = x4[i];
    float a[4] = {v.x, v.y, v.z, v.w};
    #pragma unroll
    for (int k = 0; k < 4; ++k){
      float f = a[k];
      s += (double)f;
      q += (double)f * (double)f;
      mn = fminf(mn, f); mx = fmaxf(mx, f);
      atomicAdd(&lh[f2key(f) >> 21], 1u);
    }
  }
  sS[tid] = s; sQ[tid] = q; sMn[tid] = mn; sMx[tid] = mx;
  __syncthreads();
  for (int off = TPB/2; off > 0; off >>= 1){
    if (tid < off){
      sS[tid] += sS[tid+off]; sQ[tid] += sQ[tid+off];
      sMn[tid] = fminf(sMn[tid], sMn[tid+off]);
      sMx[tid] = fmaxf(sMx[tid], sMx[tid+off]);
    }
    __syncthreads();
  }
  if (tid == 0){
    ((double*)(ws + OFF_PSUM))[blockIdx.x] = sS[0];
    ((double*)(ws + OFF_PSQ ))[blockIdx.x] = sQ[0];
    Scalars* sc = (Scalars*)(ws + OFF_SCAL);
    atomicMin(&sc->minKey, f2key(sMn[0]));
    atomicMax(&sc->maxKey, f2key(sMx[0]));
  }
  unsigned* gh = (unsigned*)(ws + OFF_H1);
  for (int i = tid; i < 2048; i += TPB){
    unsigned c = lh[i];
    if (c) atomicAdd(&gh[i], c);
  }
}

// ---------------- K2: finalize std / coeff ----------------
__global__ void __launch_bounds__(TPB) k_finalize(char* __restrict__ ws){
  __shared__ double s1[TPB], s2[TPB];
  const int tid = threadIdx.x;
  const double* psum = (const double*)(ws + OFF_PSUM);
  const double* psq  = (const double*)(ws + OFF_PSQ);
  double a = 0.0, b = 0.0;
  for (int i = tid; i < GRID1; i += TPB){ a += psum[i]; b += psq[i]; }
  s1[tid] = a; s2[tid] = b;
  __syncthreads();
  for (int off = TPB/2; off > 0; off >>= 1){
    if (tid < off){ s1[tid] += s1[tid+off]; s2[tid] += s2[tid+off]; }
    __syncthreads();
  }
  if (tid == 0){
    Scalars* sc = (Scalars*)(ws + OFF_SCAL);
    double S = s1[0], Q = s2[0];
    double n = (double)NTOTAL;
    double var = (Q - S * S / n) / (n - 1.0);
    float stdv = (float)sqrt(var);
    float fmn = key2f(sc->minKey), fmx = key2f(sc->maxKey);
    float coeff = stdv / (fmx - fmn);
    sc->stdv = stdv; sc->coeff = coeff; sc->bandC = coeff * 0.8f;
  }
}

// ---------------- radix-select scan step (serial over <=2048 bins) ----------------
__global__ void k_scan(char* __restrict__ ws, int histOff, int nbins, int binBits, int last){
  if (threadIdx.x != 0) return;
  Scalars* sc = (Scalars*)(ws + OFF_SCAL);
  const unsigned* h = (const unsigned*)(ws + histOff);
  unsigned long long k = sc->kRemain, cum = 0ULL;
  unsigned sel = 0u;
  for (int i = 0; i < nbins; ++i){
    unsigned long long c = cum + (unsigned long long)h[i];
    if (k < c){ sel = (unsigned)i; break; }
    cum = c;
  }
  sc->kRemain = k - cum;
  sc->selPrefix = (sc->selPrefix << binBits) | sel;
  if (last) sc->mean = key2f(sc->selPrefix);
}

// ---------------- histogram refinement pass ----------------
__global__ void __launch_bounds__(TPB) k_hist_refine(const float* __restrict__ x, char* __restrict__ ws,
                                                     int histOff, int shift, int binBits){
  const Scalars* sc = (const Scalars*)(ws + OFF_SCAL);
  const unsigned target = sc->selPrefix;
  const int nb = 1 << binBits;
  __shared__ unsigned lh[2048];
  const int tid = threadIdx.x;
  for (int i = tid; i < nb; i += TPB) lh[i] = 0u;
  __syncthreads();
  const float4* x4 = (const float4*)x;
  const long long n4 = NTOTAL >> 2;
  for (long long i = (long long)blockIdx.x * TPB + tid; i < n4; i += (long long)gridDim.x * TPB){
    float4 v = x4[i];
    float a[4] = {v.x, v.y, v.z, v.w};
    #pragma unroll
    for (int k = 0; k < 4; ++k){
      unsigned key = f2key(a[k]);
      if ((key >> shift) == target)
        atomicAdd(&lh[(key >> (shift - binBits)) & (unsigned)(nb - 1)], 1u);
    }
  }
  __syncthreads();
  unsigned* gh = (unsigned*)(ws + histOff);
  for (int i = tid; i < nb; i += TPB){
    unsigned c = lh[i];
    if (c) atomicAdd(&gh[i], c);
  }
}

// ---------------- K7: per-(b,t) row stats, row staged to LDS via TDM ----------------
__global__ void __launch_bounds__(TPB) k_rowstats(const float* __restrict__ x, char* __restrict__ ws){
  const int row = blockIdx.x;            // b*2048 + t
  const int b = row >> 11, t = row & 2047;
  const Scalars* sc = (const Scalars*)(ws + OFF_SCAL);
  __shared__ float rowv[CROW];
  __shared__ float sA[TPB], sB[TPB];
  __shared__ int   sI[TPB];
  __shared__ float sPar[3];              // err, hi, lo
  const int tid = threadIdx.x;

  if (tid < 32){                         // one wave issues the TDM descriptor
    uint32_t lds = (uint32_t)(uintptr_t)(&rowv[0]);
    const float* g = x + ((uint64_t)b * NH * NT * NC + (uint64_t)t * NC);
    tdm_load_tile_f32(lds, g, NC, NH, (uint64_t)NT * NC);  // 16 rows x 128 f32, stride t*c
  }
  __syncthreads();

  float mn = __int_as_float(0x7f800000), mx = -mn;
  for (int j = tid; j < CROW; j += TPB){ float v = rowv[j]; mn = fminf(mn, v); mx = fmaxf(mx, v); }
  sA[tid] = mn; sB[tid] = mx;
  __syncthreads();
  for (int off = TPB/2; off > 0; off >>= 1){
    if (tid < off){ sA[tid] = fminf(sA[tid], sA[tid+off]); sB[tid] = fmaxf(sB[tid], sB[tid+off]); }
    __syncthreads();
  }
  if (tid == 0){
    float rmin = sA[0], rmax = sB[0];
    float err = rmax - rmin;
    float band = err * sc->bandC;
    sPar[0] = err; sPar[1] = sc->mean + band; sPar[2] = sc->mean - band;
    ((float*)(ws + OFF_RMIN))[row] = rmin;
    ((float*)(ws + OFF_RMAX))[row] = rmax;
  }
  __syncthreads();
  const float hi = sPar[1], lo = sPar[2];
  int cnt = 0;
  for (int j = tid; j < CROW; j += TPB){ float v = rowv[j]; cnt += (v < hi) ? 1 : 0; cnt -= (v < lo) ? 1 : 0; }
  sI[tid] = cnt;
  __syncthreads();
  for (int off = TPB/2; off > 0; off >>= 1){
    if (tid < off) sI[tid] += sI[tid+off];
    __syncthreads();
  }
  if (tid == 0){
    float a = fmaxf((float)sI[0], CNT_FLOOR);
    a = 1.0f / a;
    if (a > 1.0f) a = 0.0f;              // dead branch, kept for exact semantics
    ((float*)(ws + OFF_RERR ))[row] = sPar[0];
    ((float*)(ws + OFF_RASTD))[row] = a;
  }
}

// ---------------- K8: per-batch rank / min_idx / bit assignment ----------------
__global__ void __launch_bounds__(1024) k_batch(char* __restrict__ ws){
  const int b = blockIdx.x, tid = threadIdx.x;
  __shared__ float errv[NT], astv[NT], dE[NT], dA[NT];
  __shared__ int sMin;
  const float* rE = (const float*)(ws + OFF_RERR ) + b * NT;
  const float* rA = (const float*)(ws + OFF_RASTD) + b * NT;
  for (int i = tid; i < NT; i += 1024){ errv[i] = rE[i]; astv[i] = rA[i]; }
  if (tid == 0) sMin = NT;
  __syncthreads();

  int rk[2];
  #pragma unroll
  for (int kk = 0; kk < 2; ++kk){
    const int i = tid + kk * 1024;
    const float e = errv[i];
    int r = 0;
    for (int j = 0; j < NT; ++j){
      float ej = errv[j];
      r += ((ej > e) || (ej == e && j < i)) ? 1 : 0;   // stable descending rank
    }
    rk[kk] = r;
    dE[r] = e;        // scatter sorted-desc err
    dA[r] = astv[i];  // attn_std aligned with dec order
  }
  __syncthreads();

  const Scalars* sc = (const Scalars*)(ws + OFF_SCAL);
  const float stdv = sc->stdv;
  #pragma unroll
  for (int kk = 0; kk < 2; ++kk){
    const int r = tid + kk * 1024;
    float s = (dE[r] * dA[r] - dE[NT-1-r] * dA[NT-1-r] * THRESH) / stdv;
    if (!(s > 0.0f)) atomicMin(&sMin, r);
  }
  __syncthreads();
  const int mi = (sMin == NT) ? -1 : sMin;
  int* rB = (int*)(ws + OFF_RBIT) + b * NT;
  #pragma unroll
  for (int kk = 0; kk < 2; ++kk){
    const int i = tid + kk * 1024, r = rk[kk];
    rB[i] = (r < mi) ? 9 : ((r > NT - 1 - mi) ? 7 : 8);
  }
}

// ---------------- K9: dequantize pass ----------------
__device__ __forceinline__ float dq1(float v, float delta, float zp, float nm1){
  float xi = rintf(v / delta) + zp;          // round_ste + zero_point (RNE like jnp.round)
  float q = xi / nm1;
  q = fminf(fmaxf(q, 0.0f), 1.0f);
  float xq = q * nm1;
  return (xq - zp) * delta;
}

__global__ void __launch_bounds__(TPB) k_dequant(const float* __restrict__ x, float* __restrict__ out,
                                                 const char* __restrict__ ws){
  const int row = blockIdx.x;
  const int b = row >> 11, t = row & 2047;
  const float xmin = ((const float*)(ws + OFF_RMIN))[row];
  const float xmax = ((const float*)(ws + OFF_RMAX))[row];
  const int bit = ((const int*)(ws + OFF_RBIT))[row];
  const float nl = (float)(1 << bit);
  const float nm1 = nl - 1.0f;
  float delta = (xmax - xmin) / nm1;
  if (delta < 1e-8f) delta = 1e-8f;
  const float zp = rintf(-xmin / delta);

  const uint64_t base4 = ((uint64_t)b * NH * NT * NC + (uint64_t)t * NC) >> 2;  // in float4 units
  const float4* x4 = (const float4*)x;
  float4* o4 = (float4*)out;
  for (int j4 = threadIdx.x; j4 < CROW/4; j4 += TPB){
    const int h = j4 >> 5;                      // 32 float4 per h-slab
    const int c4 = j4 & 31;
    const uint64_t idx = base4 + (uint64_t)h * ((NT * NC) >> 2) + (uint64_t)c4;
    float4 v = x4[idx];
    float4 r;
    r.x = dq1(v.x, delta, zp, nm1);
    r.y = dq1(v.y, delta, zp, nm1);
    r.z = dq1(v.z, delta, zp, nm1);
    r.w = dq1(v.w, delta, zp, nm1);
    o4[idx] = r;
  }
}

// ---------------- launch ----------------
extern "C" void kernel_launch(void* const* d_in, const int* in_sizes, int n_in,
                              void* d_out, int out_size, void* d_ws, size_t ws_size,
                              hipStream_t stream) {
  const float* x = (const float*)d_in[0];
  float* out = (float*)d_out;
  char* ws = (char*)d_ws;

  k_init<<<20, 256, 0, stream>>>(ws);
  k_stats_hist1<<<GRID1, TPB, 0, stream>>>(x, ws);
  k_finalize<<<1, TPB, 0, stream>>>(ws);
  k_scan<<<1, 32, 0, stream>>>(ws, OFF_H1, 2048, 11, 0);
  k_hist_refine<<<GRID1, TPB, 0, stream>>>(x, ws, OFF_H2, 21, 11);
  k_scan<<<1, 32, 0, stream>>>(ws, OFF_H2, 2048, 11, 0);
  k_hist_refine<<<GRID1, TPB, 0, stream>>>(x, ws, OFF_H3, 10, 10);
  k_scan<<<1, 32, 0, stream>>>(ws, OFF_H3, 1024, 10, 1);
  k_rowstats<<<NROWS, TPB, 0, stream>>>(x, ws);
  k_batch<<<NB, 1024, 0, stream>>>(ws);
  k_dequant<<<NROWS, TPB, 0, stream>>>(x, out, ws);
}